// HCA_GCN_24060406792215
// MI455X (gfx1250) — compile-verified
//
#include <hip/hip_runtime.h>

typedef __attribute__((ext_vector_type(2))) float v2f;
typedef __attribute__((ext_vector_type(8))) float v8f;

#define NNODES 10000
#define FDIM   128
#define LDSP   132   // padded LDS row stride (floats) to avoid bank conflicts

// C[M x 128] = A[M x K] (row-major, lda) @ B[K x 128], optional bias + relu.
// One block: 32 output rows x 128 output cols. 8 waves; wave w -> cols 16w..16w+15,
// each wave keeps two stacked 16x16 f32 accumulators (rows 0-15 and 16-31 of the
// block tile) that share one B fragment per V_WMMA_F32_16X16X4_F32 k-step.
// Requires K % 4 == 0 (true here: 128 and 10000).
__global__ __launch_bounds__(256)
void gcn_gemm_wmma(const float* __restrict__ A, int lda, int M, int K,
                   const float* __restrict__ B,
                   const float* __restrict__ bias, int relu,
                   float* __restrict__ C)
{
    __shared__ float As[32 * LDSP];   // 16.5 KB

    const int tid   = threadIdx.x;
    const int lane  = tid & 31;
    const int wave  = tid >> 5;
    const int row0  = blockIdx.x * 32;
    const int colb  = wave * 16;

    const int m16   = lane & 15;          // A-frag row / B-frag col within tile
    const int khalf = (lane >> 4) << 1;   // K sub-offset: lanes 0-15 -> 0, lanes 16-31 -> 2

    v8f acc0 = {};
    v8f acc1 = {};

    for (int k0 = 0; k0 < K; k0 += 128) {
        const int krem   = K - k0;
        const int ksteps = krem < 128 ? krem : 128;   // multiple of 4 by construction

        // ---- cooperative A-panel load: 32 rows x 128 K (1024 float4 loads) ----
        // Row index clamped so the last (partial-M) block stays in bounds; the
        // duplicated rows feed acc1 lanes whose stores are guarded off below.
        if (krem >= 128) {
            #pragma unroll
            for (int i = tid; i < 1024; i += 256) {
                const int r    = i >> 5;
                const int c4   = (i & 31) << 2;
                const int grow = (row0 + r < M) ? (row0 + r) : (M - 1);
                const float* src = A + (size_t)grow * lda + k0 + c4;
                *(float4*)(&As[r * LDSP + c4]) = *(const float4*)src;
                if (k0 + 128 < K)
                    __builtin_prefetch(src + 128, 0, 1);   // global_prefetch_b8: next panel
            }
        } else {
            #pragma unroll
            for (int i = tid; i < 1024; i += 256) {
                const int r    = i >> 5;
                const int c4   = (i & 31) << 2;
                const int grow = (row0 + r < M) ? (row0 + r) : (M - 1);
                float4 v = make_float4(0.f, 0.f, 0.f, 0.f);
                if (c4 + 4 <= krem)
                    v = *(const float4*)(A + (size_t)grow * lda + k0 + c4);
                *(float4*)(&As[r * LDSP + c4]) = v;
            }
        }
        __syncthreads();

        // ---- unguarded WMMA loop (k0 + kk + khalf + 1 < K always holds) ----
        const float* bp = B + (size_t)(k0 + khalf) * FDIM + colb + m16;
        #pragma unroll 8
        for (int kk = 0; kk < ksteps; kk += 4) {
            const int ko = kk + khalf;
            v2f a0 = *(const v2f*)(&As[m16 * LDSP + ko]);          // ds_load_b64
            v2f a1 = *(const v2f*)(&As[(16 + m16) * LDSP + ko]);   // ds_load_b64
            v2f b;
            b.x = bp[0];
            b.y = bp[FDIM];
            bp += 4 * FDIM;
            acc0 = __builtin_amdgcn_wmma_f32_16x16x4_f32(
                     false, a0, false, b, (short)0, acc0, false, false);
            acc1 = __builtin_amdgcn_wmma_f32_16x16x4_f32(
                     false, a1, false, b, (short)0, acc1, false, false);
        }
        __syncthreads();
    }

    // ---- epilogue: D layout = VGPR i -> row i (lanes 0-15) / row i+8 (lanes 16-31) ----
    const int col   = colb + m16;
    const float bv  = bias ? bias[col] : 0.f;
    const int rhalf = (lane >> 4) << 3;
    #pragma unroll
    for (int i = 0; i < 8; ++i) {
        const int r0 = row0 + i + rhalf;        // acc0 tile: block rows 0-15
        const int r1 = r0 + 16;                 // acc1 tile: block rows 16-31
        float v0 = acc0[i] + bv;
        float v1 = acc1[i] + bv;
        if (relu) { v0 = fmaxf(v0, 0.f); v1 = fmaxf(v1, 0.f); }
        if (r0 < M) C[(size_t)r0 * FDIM + col] = v0;
        if (r1 < M) C[(size_t)r1 * FDIM + col] = v1;
    }
}

// out[row] = dot(h[row,:], W2) + b2 : one wave32 per row, float4 per lane.
__global__ __launch_bounds__(256)
void gcn_readout(const float* __restrict__ H, const float* __restrict__ W2,
                 const float* __restrict__ b2, float* __restrict__ out, int M)
{
    const int lane = threadIdx.x & 31;
    const int row  = blockIdx.x * 8 + (threadIdx.x >> 5);
    if (row >= M) return;

    const float4 h4 = *(const float4*)(H + (size_t)row * FDIM + lane * 4);
    const float4 w4 = *(const float4*)(W2 + lane * 4);
    float v = h4.x * w4.x + h4.y * w4.y + h4.z * w4.z + h4.w * w4.w;

    #pragma unroll
    for (int off = 16; off > 0; off >>= 1)
        v += __shfl_xor(v, off, 32);

    if (lane == 0) out[row] = v + b2[0];
}

extern "C" void kernel_launch(void* const* d_in, const int* in_sizes, int n_in,
                              void* d_out, int out_size, void* d_ws, size_t ws_size,
                              hipStream_t stream) {
    const float* x  = (const float*)d_in[0];   // [10000,128]
    const float* a  = (const float*)d_in[1];   // [10000,10000]
    const float* W1 = (const float*)d_in[2];   // [128,128]
    const float* b1 = (const float*)d_in[3];   // [128]
    const float* W2 = (const float*)d_in[4];   // [128,1]
    const float* b2 = (const float*)d_in[5];   // [1]
    float* out = (float*)d_out;                // [10000]

    float* xw = (float*)d_ws;                  // [10000,128]
    float* h  = xw + (size_t)NNODES * FDIM;    // [10000,128]

    dim3 blk(256);
    const int gridM = (NNODES + 31) / 32;      // 313

    // Stage 1: xw = x @ W1             (K = 128, no bias/relu)
    gcn_gemm_wmma<<<gridM, blk, 0, stream>>>(x, FDIM, NNODES, FDIM, W1, nullptr, 0, xw);

    // Stage 2: h = relu(a @ xw + b1)   (K = 10000, the 400MB HBM stream)
    gcn_gemm_wmma<<<gridM, blk, 0, stream>>>(a, NNODES, NNODES, NNODES, xw, b1, 1, h);

    // Stage 3: out = h @ W2 + b2
    gcn_readout<<<(NNODES + 7) / 8, blk, 0, stream>>>(h, W2, b2, out, NNODES);
}